// SpatialGAT_16432544874816
// MI455X (gfx1250) — compile-verified
//
#include <hip/hip_runtime.h>
#include <hip/hip_bf16.h>
#include <math.h>

typedef __attribute__((ext_vector_type(2))) float v2f;
typedef __attribute__((ext_vector_type(8))) float v8f;

#define NEG_SLOPE 0.2f
#define EPS_F 1e-16f

// ---------- order-preserving float<->uint key for atomic segment-max ----------
__device__ __forceinline__ unsigned fkey(float f) {
  unsigned u = __float_as_uint(f);
  return (u & 0x80000000u) ? ~u : (u | 0x80000000u);
}
__device__ __forceinline__ float funkey(unsigned k) {
  unsigned u = (k & 0x80000000u) ? (k & 0x7FFFFFFFu) : ~k;
  return __uint_as_float(u);
}

// ---------- fills ----------
__global__ void fill_f32(float* p, float v, int n) {
  int i = blockIdx.x * blockDim.x + threadIdx.x;
  if (i < n) p[i] = v;
}
__global__ void fill_u32(unsigned* p, unsigned v, int n) {
  int i = blockIdx.x * blockDim.x + threadIdx.x;
  if (i < n) p[i] = v;
}

// ---------- fp32 GEMM via V_WMMA_F32_16X16X4_F32, one wave per 16x16 C tile ----------
// A: [M,K] row-major, B: [K,Ncols] row-major, C: [M,Ncols] row-major.
__global__ void gemm_f32_wmma(const float* __restrict__ A, const float* __restrict__ B,
                              float* __restrict__ C, int M, int K, int Ncols) {
  const int lane = threadIdx.x;      // 32 threads per block (one wave)
  const int m    = lane & 15;        // matrix row within tile (A) / col (B,C)
  const int kh   = lane >> 4;        // lane-half selects K pair (A,B) / +8 rows (C)
  const int tm   = blockIdx.x;
  const int tn   = blockIdx.y;

  int arow_idx = tm * 16 + m;
  if (arow_idx >= M) arow_idx = M - 1;          // clamp (EXEC must stay all-1 for WMMA)
  const float* arow = A + (long long)arow_idx * K;
  const float* bcol = B + tn * 16 + m;

  v8f acc = {};
  for (int kb = 0; kb < K; kb += 4) {
    const int k0 = kb + 2 * kh;
    v2f a, b;
    // A 16x4 layout: VGPR0 = K {0|2}, VGPR1 = K {1|3} split by lane-half
    a.x = arow[k0];
    a.y = arow[k0 + 1];
    // B 4x16 layout: VGPR0 = row K {0|2}, VGPR1 = row K {1|3}, cols across lanes
    b.x = bcol[(long long)(k0)     * Ncols];
    b.y = bcol[(long long)(k0 + 1) * Ncols];
    acc = __builtin_amdgcn_wmma_f32_16x16x4_f32(
        /*neg_a=*/false, a, /*neg_b=*/false, b,
        /*c_mod=*/(short)0, acc, /*reuse_a=*/false, /*reuse_b=*/false);
  }
  // C 16x16 layout: VGPR i -> row i (lanes 0-15) / row i+8 (lanes 16-31)
  const int crow0 = tm * 16 + kh * 8;
  float* cp = C + (long long)crow0 * Ncols + tn * 16 + m;
#pragma unroll
  for (int i = 0; i < 8; ++i)
    if (crow0 + i < M) cp[(long long)i * Ncols] = acc[i];
}

// ---------- per-(node,head) attention coefficients ----------
template <int H, int F>
__global__ void attn_coef(const float* __restrict__ hfeat,
                          const float* __restrict__ a_src, const float* __restrict__ a_dst,
                          float* __restrict__ as, float* __restrict__ ad, int N) {
  int idx = blockIdx.x * blockDim.x + threadIdx.x;   // n*H + h
  if (idx >= N * H) return;
  const int hh = idx & (H - 1);
  const float* hp = hfeat + (long long)idx * F;      // [N, H*F] contiguous
  float ss = 0.f, sd = 0.f;
#pragma unroll
  for (int f = 0; f < F; ++f) {
    float v = hp[f];
    ss = fmaf(v, a_src[hh * F + f], ss);
    sd = fmaf(v, a_dst[hh * F + f], sd);
  }
  as[idx] = ss;
  ad[idx] = sd;
}

// edge_index is [2,E] int32 row-major; edges e>=E are the appended self-loops.
__device__ __forceinline__ void edge_ep(const int* __restrict__ ei, int E, int e,
                                        int& s, int& d) {
  if (e < E) { s = ei[e]; d = ei[E + e]; }
  else       { s = d = e - E; }
}

template <int H>
__global__ void edge_logit_max(const int* __restrict__ ei,
                               const float* __restrict__ as, const float* __restrict__ ad,
                               unsigned* __restrict__ emax, int E, int Etot) {
  int idx = blockIdx.x * blockDim.x + threadIdx.x;   // e*H + h
  if (idx >= Etot * H) return;
  const int e = idx / H, hh = idx - e * H;
  int s, d; edge_ep(ei, E, e, s, d);
  float v = as[s * H + hh] + ad[d * H + hh];
  v = v > 0.f ? v : NEG_SLOPE * v;
  atomicMax(&emax[d * H + hh], fkey(v));
}

template <int H>
__global__ void edge_exp_sum(const int* __restrict__ ei,
                             const float* __restrict__ as, const float* __restrict__ ad,
                             const unsigned* __restrict__ emax,
                             float* __restrict__ exbuf, float* __restrict__ denom,
                             int E, int Etot) {
  int idx = blockIdx.x * blockDim.x + threadIdx.x;   // e*H + h
  if (idx >= Etot * H) return;
  const int e = idx / H, hh = idx - e * H;
  int s, d; edge_ep(ei, E, e, s, d);
  float v = as[s * H + hh] + ad[d * H + hh];
  v = v > 0.f ? v : NEG_SLOPE * v;
  unsigned k = emax[d * H + hh];
  float m = (k == 0u) ? 0.f : funkey(k);             // mirrors isfinite->0 guard
  float ex = expf(v - m);
  exbuf[idx] = ex;
  atomicAdd(&denom[d * H + hh], ex);
}

// thread = (edge, channel); consecutive lanes cover consecutive features of h[src]
template <int H, int F>
__global__ void edge_aggregate(const int* __restrict__ ei, const float* __restrict__ hfeat,
                               const float* __restrict__ exbuf, const float* __restrict__ denom,
                               float* __restrict__ out, int E, int Etot) {
  const int HF = H * F;
  long long idx = (long long)blockIdx.x * blockDim.x + threadIdx.x;
  if (idx >= (long long)Etot * HF) return;
  const int e  = (int)(idx / HF);
  const int c  = (int)(idx - (long long)e * HF);
  const int hh = c / F;
  int s, d; edge_ep(ei, E, e, s, d);
  const float alpha = exbuf[e * H + hh] / (denom[d * H + hh] + EPS_F);
  atomicAdd(&out[(long long)d * HF + c], alpha * hfeat[(long long)s * HF + c]);
}

__global__ void bias_relu_inplace(float* __restrict__ p, const float* __restrict__ b,
                                  int n, int cmask) {
  int i = blockIdx.x * blockDim.x + threadIdx.x;
  if (i >= n) return;
  float v = p[i] + b[i & cmask];
  p[i] = v > 0.f ? v : 0.f;
}
__global__ void bias_add_inplace(float* __restrict__ p, const float* __restrict__ b,
                                 int n, int cmask) {
  int i = blockIdx.x * blockDim.x + threadIdx.x;
  if (i >= n) return;
  p[i] = p[i] + b[i & cmask];
}

static inline int cdiv(long long a, int b) { return (int)((a + b - 1) / b); }

extern "C" void kernel_launch(void* const* d_in, const int* in_sizes, int n_in,
                              void* d_out, int out_size, void* d_ws, size_t ws_size,
                              hipStream_t stream) {
  const float* x      = (const float*)d_in[0];
  const int*   ei     = (const int*)  d_in[1];
  const float* W1     = (const float*)d_in[2];
  const float* a_src1 = (const float*)d_in[3];
  const float* a_dst1 = (const float*)d_in[4];
  const float* b1     = (const float*)d_in[5];
  const float* W2     = (const float*)d_in[6];
  const float* a_src2 = (const float*)d_in[7];
  const float* a_dst2 = (const float*)d_in[8];
  const float* b2     = (const float*)d_in[9];

  const int IN = 64, HID = 32;          // HEADS=2 layer1, 1 layer2
  const int N    = in_sizes[0] / IN;    // 50000
  const int E    = in_sizes[1] / 2;     // 800000
  const int Etot = E + N;               // self-loops appended

  // ---- workspace carve ----
  float* ws = (float*)d_ws;
  float*    h1   = ws;               ws += (size_t)N * 64;   // x@W1
  float*    out1 = ws;               ws += (size_t)N * 64;   // layer1 agg -> relu(.+b1)
  float*    h2   = ws;               ws += (size_t)N * 32;   // out1@W2
  float*    as1  = ws;               ws += (size_t)N * 2;
  float*    ad1  = ws;               ws += (size_t)N * 2;
  float*    as2  = ws;               ws += (size_t)N;
  float*    ad2  = ws;               ws += (size_t)N;
  float*    den1 = ws;               ws += (size_t)N * 2;
  float*    den2 = ws;               ws += (size_t)N;
  unsigned* em1  = (unsigned*)ws;    ws += (size_t)N * 2;
  unsigned* em2  = (unsigned*)ws;    ws += (size_t)N;
  float*    ex1  = ws;               ws += (size_t)Etot * 2;
  float*    ex2  = ws;               ws += (size_t)Etot;
  float*    outp = (float*)d_out;

  const int B = 256;

  // ================= layer 1 (H=2, F=32) =================
  fill_f32<<<cdiv((long long)N * 64, B), B, 0, stream>>>(out1, 0.f, N * 64);
  fill_f32<<<cdiv(N * 2, B), B, 0, stream>>>(den1, 0.f, N * 2);
  fill_u32<<<cdiv(N * 2, B), B, 0, stream>>>(em1, 0u, N * 2);

  {
    dim3 grid(cdiv(N, 16), 64 / 16);   // 3125 x 4 tiles, 1 wave each
    gemm_f32_wmma<<<grid, 32, 0, stream>>>(x, W1, h1, N, IN, 64);
  }
  attn_coef<2, 32><<<cdiv(N * 2, B), B, 0, stream>>>(h1, a_src1, a_dst1, as1, ad1, N);
  edge_logit_max<2><<<cdiv((long long)Etot * 2, B), B, 0, stream>>>(ei, as1, ad1, em1, E, Etot);
  edge_exp_sum<2><<<cdiv((long long)Etot * 2, B), B, 0, stream>>>(ei, as1, ad1, em1, ex1, den1, E, Etot);
  edge_aggregate<2, 32><<<cdiv((long long)Etot * 64, B), B, 0, stream>>>(ei, h1, ex1, den1, out1, E, Etot);
  bias_relu_inplace<<<cdiv((long long)N * 64, B), B, 0, stream>>>(out1, b1, N * 64, 63);

  // ================= layer 2 (H=1, F=32) =================
  fill_f32<<<cdiv((long long)N * 32, B), B, 0, stream>>>(outp, 0.f, N * 32);
  fill_f32<<<cdiv(N, B), B, 0, stream>>>(den2, 0.f, N);
  fill_u32<<<cdiv(N, B), B, 0, stream>>>(em2, 0u, N);

  {
    dim3 grid(cdiv(N, 16), 32 / 16);   // 3125 x 2 tiles
    gemm_f32_wmma<<<grid, 32, 0, stream>>>(out1, W2, h2, N, 64, 32);
  }
  attn_coef<1, 32><<<cdiv(N, B), B, 0, stream>>>(h2, a_src2, a_dst2, as2, ad2, N);
  edge_logit_max<1><<<cdiv(Etot, B), B, 0, stream>>>(ei, as2, ad2, em2, E, Etot);
  edge_exp_sum<1><<<cdiv(Etot, B), B, 0, stream>>>(ei, as2, ad2, em2, ex2, den2, E, Etot);
  edge_aggregate<1, 32><<<cdiv((long long)Etot * 32, B), B, 0, stream>>>(ei, h2, ex2, den2, outp, E, Etot);
  bias_add_inplace<<<cdiv((long long)N * 32, B), B, 0, stream>>>(outp, b2, N * 32, 31);
}